// GNN_29695403884613
// MI455X (gfx1250) — compile-verified
//
#include <hip/hip_runtime.h>

#define N_NODES 100000
#define HIDDEN  128
#define N_EDGES 1000000
#define ASTRIDE 132   // LDS row stride (floats): 128 + 4 pad -> conflict-free ds_load_b64

typedef __attribute__((ext_vector_type(2))) float v2f;
typedef __attribute__((ext_vector_type(8))) float v8f;

// ---------------------------------------------------------------------------
// Zero a float buffer (sums / counts in workspace).
// ---------------------------------------------------------------------------
__global__ void gnn_zero_kernel(float* __restrict__ p, long n) {
  long i = (long)blockIdx.x * blockDim.x + threadIdx.x;
  if (i < n) p[i] = 0.0f;
}

// ---------------------------------------------------------------------------
// Edge scatter: one wave (32 lanes) per edge; lane handles 4 channels.
// sums[dst] += x[src]; cnt[dst] += 1 (first layer only).
// ---------------------------------------------------------------------------
__global__ __launch_bounds__(256) void gnn_scatter_kernel(
    const float* __restrict__ x,
    const long long* __restrict__ src,
    const long long* __restrict__ dst,
    float* __restrict__ sums,
    float* __restrict__ cnt,
    int do_count) {
  long tid = (long)blockIdx.x * blockDim.x + threadIdx.x;
  int e    = (int)(tid >> 5);
  int lane = (int)(tid & 31);
  if (e >= N_EDGES) return;
  int s = (int)src[e];
  int d = (int)dst[e];
  const float4 v = *(const float4*)(x + (size_t)s * HIDDEN + lane * 4);
  float* p = sums + (size_t)d * HIDDEN + lane * 4;
  atomicAdd(p + 0, v.x);
  atomicAdd(p + 1, v.y);
  atomicAdd(p + 2, v.z);
  atomicAdd(p + 3, v.w);
  if (do_count && lane == 0) atomicAdd(cnt + d, 1.0f);
}

// ---------------------------------------------------------------------------
// Fused SAGE layer: out = [sums/max(cnt,1)] @ w_l + b_l + x @ w_r (+ ReLU)
// One block = 16 output rows; 8 waves = 8 column tiles of 16x16.
// fp32 WMMA: V_WMMA_F32_16X16X4_F32, K-loop 128/4 = 32 steps per operand pair.
// ---------------------------------------------------------------------------
__global__ __launch_bounds__(256) void gnn_sage_gemm_kernel(
    const float* __restrict__ sums, const float* __restrict__ cnt,
    const float* __restrict__ x,
    const float* __restrict__ wl, const float* __restrict__ bl,
    const float* __restrict__ wr,
    float* __restrict__ out, int relu) {
  __shared__ float sA[16 * ASTRIDE];  // mean-aggregated tile (division fused)
  __shared__ float sX[16 * ASTRIDE];  // raw feature tile

  const int row_base = blockIdx.x * 16;

  // Cooperative, coalesced staging: 16 rows x 128 cols = 512 float4 per matrix.
  for (int f = threadIdx.x; f < 512; f += 256) {
    int r = f >> 5;
    int c = (f & 31) << 2;
    int grow = row_base + r;
    float inv = 1.0f / fmaxf(cnt[grow], 1.0f);
    float4 v = *(const float4*)(sums + (size_t)grow * HIDDEN + c);
    float4 u = *(const float4*)(x    + (size_t)grow * HIDDEN + c);
    float* pa = sA + r * ASTRIDE + c;
    float* px = sX + r * ASTRIDE + c;
    pa[0] = v.x * inv; pa[1] = v.y * inv; pa[2] = v.z * inv; pa[3] = v.w * inv;
    px[0] = u.x;       px[1] = u.y;       px[2] = u.z;       px[3] = u.w;
  }
  __syncthreads();

  const int lane = threadIdx.x & 31;
  const int wv   = threadIdx.x >> 5;   // column tile 0..7
  const int l15  = lane & 15;          // A row / D column within tile
  const int half = lane >> 4;          // 0: K=k0,k0+1   1: K=k0+2,k0+3
  const int col  = wv * 16 + l15;      // global output column

  v8f acc = {};

  // aggr @ w_l
#pragma unroll
  for (int kk = 0; kk < 32; ++kk) {
    int k = kk * 4 + half * 2;
    v2f a = *(const v2f*)(&sA[l15 * ASTRIDE + k]);        // ds_load_b64, conflict-free
    v2f b;                                                 // lanes 0-15: one cacheline each
    b.x = wl[(size_t)k * HIDDEN + col];
    b.y = wl[(size_t)(k + 1) * HIDDEN + col];
    acc = __builtin_amdgcn_wmma_f32_16x16x4_f32(false, a, false, b,
                                                (short)0, acc, false, false);
  }
  // x @ w_r
#pragma unroll
  for (int kk = 0; kk < 32; ++kk) {
    int k = kk * 4 + half * 2;
    v2f a = *(const v2f*)(&sX[l15 * ASTRIDE + k]);
    v2f b;
    b.x = wr[(size_t)k * HIDDEN + col];
    b.y = wr[(size_t)(k + 1) * HIDDEN + col];
    acc = __builtin_amdgcn_wmma_f32_16x16x4_f32(false, a, false, b,
                                                (short)0, acc, false, false);
  }

  // Bias + optional ReLU + store. D layout: VGPR i -> M=i (lanes 0-15), M=8+i (16-31).
  const float bias = bl[col];
#pragma unroll
  for (int i = 0; i < 8; ++i) {
    float r = acc[i] + bias;
    if (relu) r = fmaxf(r, 0.0f);
    out[(size_t)(row_base + half * 8 + i) * HIDDEN + col] = r;
  }
}

// ---------------------------------------------------------------------------
// Launch: two SAGE layers. Degree counts computed once (graph is static).
// Workspace: sums [N*D] | cnt [N] | h1 [N*D]   (~103 MB)
// ---------------------------------------------------------------------------
extern "C" void kernel_launch(void* const* d_in, const int* in_sizes, int n_in,
                              void* d_out, int out_size, void* d_ws, size_t ws_size,
                              hipStream_t stream) {
  (void)in_sizes; (void)n_in; (void)out_size; (void)ws_size;

  const long long* ei  = (const long long*)d_in[0];
  const long long* src = ei;
  const long long* dst = ei + N_EDGES;
  const float* x0  = (const float*)d_in[1];
  const float* w1l = (const float*)d_in[2];
  const float* b1l = (const float*)d_in[3];
  const float* w1r = (const float*)d_in[4];
  const float* w2l = (const float*)d_in[5];
  const float* b2l = (const float*)d_in[6];
  const float* w2r = (const float*)d_in[7];
  float* out = (float*)d_out;

  float* sums = (float*)d_ws;
  float* cnt  = sums + (size_t)N_NODES * HIDDEN;
  float* h1   = cnt + N_NODES;

  const long nz   = (long)N_NODES * HIDDEN + N_NODES;  // sums + cnt contiguous
  const long ns   = (long)N_NODES * HIDDEN;
  const long nthr = (long)N_EDGES * 32;
  const unsigned zb  = (unsigned)((nz + 255) / 256);
  const unsigned sb  = (unsigned)((ns + 255) / 256);
  const unsigned eb  = (unsigned)((nthr + 255) / 256);
  const unsigned gb  = (unsigned)(N_NODES / 16);       // 6250, exact

  // Layer 1
  gnn_zero_kernel<<<zb, 256, 0, stream>>>(sums, nz);
  gnn_scatter_kernel<<<eb, 256, 0, stream>>>(x0, src, dst, sums, cnt, 1);
  gnn_sage_gemm_kernel<<<gb, 256, 0, stream>>>(sums, cnt, x0, w1l, b1l, w1r, h1, 1);

  // Layer 2 (reuse cnt)
  gnn_zero_kernel<<<sb, 256, 0, stream>>>(sums, ns);
  gnn_scatter_kernel<<<eb, 256, 0, stream>>>(h1, src, dst, sums, cnt, 0);
  gnn_sage_gemm_kernel<<<gb, 256, 0, stream>>>(sums, cnt, h1, w2l, b2l, w2r, out, 0);
}